// _ChebConvSAt_86655260164722
// MI455X (gfx1250) — compile-verified
//
#include <hip/hip_runtime.h>

typedef float v2f __attribute__((ext_vector_type(2)));
typedef float v4f __attribute__((ext_vector_type(4)));
typedef float v8f __attribute__((ext_vector_type(8)));

#define B_    32
#define N_    512
#define FIN_  64
#define FOUT_ 64
#define T_    12
#define K_    3
#define C1_   (FIN_ * T_)      // 768: flattened (f,t) columns, also (o,t) of out
#define NTILE 16               // n-rows per workgroup
#define NWAVE 16
#define THREADS (NWAVE * 32)   // 512
#define LP    772              // padded LDS pitch (floats); 772 % 64 == 4 (bank spread), % 4 == 0

// One workgroup = (b, 16-row n tile), 16 waves.
//   m-loop (all 3 k fused): acc1[k][j] (9 C tiles/wave) += A_k(16x4, cheb*att on the fly)
//                           @ B_j(4x16, x[b] rows) -- X streamed from global exactly once.
//   then per k: spill Y_k (16x768) to LDS, stage 2 accumulates acc2 (3 C tiles/wave)
//   += Y_k[:,f,t] @ Theta[k] contracting f per t.  Epilogue: relu + LDS transpose +
//   coalesced float4 stores (out written exactly once, no atomics).
__launch_bounds__(THREADS)
__global__ void cheb_attn_fused(const float* __restrict__ x,
                                const float* __restrict__ cheb,
                                const float* __restrict__ att,
                                const float* __restrict__ theta,
                                float* __restrict__ out)
{
    __shared__ __align__(16) float YL[NTILE * LP];   // ~49 KB

    const int b   = blockIdx.y;
    const int nt0 = blockIdx.x * NTILE;

    const int tid  = threadIdx.x;
    const int w    = tid >> 5;          // wave id 0..15
    const int lane = tid & 31;
    const int lo   = lane & 15;
    const int hi   = lane >> 4;         // half-wave select (0/1)

    // stage-1: wave w owns Y columns [w*48, w*48+48)  (3 tiles of 16)
    const int cwb = w * 48;
    // stage-2: wave w owns o in [ob, ob+16), t in [tb, tb+3)
    const int ob = (w & 3) * 16;
    const int tb = (w >> 2) * 3;

    // persistent stage-2 accumulators: 3 t's x one o-subtile = 3 C tiles (24 VGPRs)
    v8f acc2[3];
#pragma unroll
    for (int tt = 0; tt < 3; ++tt)
#pragma unroll
        for (int r = 0; r < 8; ++r) acc2[tt][r] = 0.0f;

    // stage-1 accumulators: 3 k x 3 column-tiles = 9 C tiles (72 VGPRs)
    v8f acc1[K_][3];
#pragma unroll
    for (int k = 0; k < K_; ++k)
#pragma unroll
        for (int j = 0; j < 3; ++j)
#pragma unroll
            for (int r = 0; r < 8; ++r) acc1[k][j][r] = 0.0f;

    const float* xb   = x   + b * N_ * C1_;             // x[b] (512 x 768 contiguous)
    const float* attN = att + b * N_ * N_ + (nt0 + lo); // fixed column n per lane
    const float* chN0 = cheb + 0 * N_ * N_ + (nt0 + lo);
    const float* chN1 = cheb + 1 * N_ * N_ + (nt0 + lo);
    const float* chN2 = cheb + 2 * N_ * N_ + (nt0 + lo);

    // ---------------- stage 1: Y_k = W_k^T @ X, contract m, all k fused ----------------
#pragma unroll 2
    for (int m0 = 0; m0 < N_; m0 += 4) {
        const int mA = m0 + 2 * hi;                 // K rows {2hi, 2hi+1} per lane
        const float a0 = attN[mA * N_];
        const float a1 = attN[(mA + 1) * N_];
        v2f A[K_];
        A[0].x = chN0[mA * N_] * a0;  A[0].y = chN0[(mA + 1) * N_] * a1;
        A[1].x = chN1[mA * N_] * a0;  A[1].y = chN1[(mA + 1) * N_] * a1;
        A[2].x = chN2[mA * N_] * a0;  A[2].y = chN2[(mA + 1) * N_] * a1;

        const float* xr0 = xb + mA * C1_ + cwb + lo;
        __builtin_prefetch(xr0 + 16 * C1_, 0, 0);   // global_prefetch_b8: stream ahead
        v2f Bv[3];
#pragma unroll
        for (int j = 0; j < 3; ++j) {
            Bv[j].x = xr0[j * 16];
            Bv[j].y = xr0[C1_ + j * 16];
        }
#pragma unroll
        for (int k = 0; k < K_; ++k)
#pragma unroll
            for (int j = 0; j < 3; ++j)
                acc1[k][j] = __builtin_amdgcn_wmma_f32_16x16x4_f32(
                    false, A[k], false, Bv[j], (short)0, acc1[k][j], false, false);
    }

    // ------------- per k: spill Y_k to LDS, stage 2 accumulates into acc2 -------------
#pragma unroll
    for (int k = 0; k < K_; ++k) {
        const float* thK = theta + k * FIN_ * FOUT_;

        // spill Y_k: C layout row = r + 8*hi, col = c0 + lo (row-contiguous stores)
#pragma unroll
        for (int j = 0; j < 3; ++j) {
            const int c0 = cwb + j * 16;
#pragma unroll
            for (int r = 0; r < 8; ++r)
                YL[(r + 8 * hi) * LP + c0 + lo] = acc1[k][j][r];
        }
        __syncthreads();

        // stage 2: contract f (64) in steps of 4
#pragma unroll 2
        for (int f0 = 0; f0 < FIN_; f0 += 4) {
            const int fA = f0 + 2 * hi;
            // B tile (Theta[k]): B[kk, o], lane = o, half-wave selects kk pair
            v2f Bt;
            Bt.x = thK[fA * FOUT_ + ob + lo];
            Bt.y = thK[(fA + 1) * FOUT_ + ob + lo];
#pragma unroll
            for (int tt = 0; tt < 3; ++tt) {
                const int t = tb + tt;
                const float* yrow = &YL[lo * LP + t];   // A[n, kk] = YL[n][(f0+kk)*12 + t]
                v2f Av;
                Av.x = yrow[fA * T_];
                Av.y = yrow[(fA + 1) * T_];
                acc2[tt] = __builtin_amdgcn_wmma_f32_16x16x4_f32(
                    false, Av, false, Bt, (short)0, acc2[tt], false, false);
            }
        }
        __syncthreads();   // YL reads done before next k's spill overwrites it
    }

    // ---------------- epilogue: transpose via LDS, relu, coalesced store ----------
#pragma unroll
    for (int tt = 0; tt < 3; ++tt) {
        const int t = tb + tt;
#pragma unroll
        for (int r = 0; r < 8; ++r)
            YL[(r + 8 * hi) * LP + (ob + lo) * T_ + t] = acc2[tt][r];
    }
    __syncthreads();

    // out[b, nt0+n, o, t]: rows of 768 contiguous floats; 32 threads per row, float4
    float* outp = out + (size_t)(b * N_ + nt0) * C1_;
    const int nl = tid >> 5;            // 0..15
    const int cb = (tid & 31) * 24;     // 32 threads x 24 floats = 768
#pragma unroll
    for (int i = 0; i < 6; ++i) {
        v4f v = *(const v4f*)&YL[nl * LP + cb + i * 4];
        v4f r;
        r.x = v.x > 0.0f ? v.x : 0.0f;
        r.y = v.y > 0.0f ? v.y : 0.0f;
        r.z = v.z > 0.0f ? v.z : 0.0f;
        r.w = v.w > 0.0f ? v.w : 0.0f;
        *(v4f*)&outp[nl * C1_ + cb + i * 4] = r;
    }
}

extern "C" void kernel_launch(void* const* d_in, const int* in_sizes, int n_in,
                              void* d_out, int out_size, void* d_ws, size_t ws_size,
                              hipStream_t stream) {
    (void)in_sizes; (void)n_in; (void)out_size; (void)d_ws; (void)ws_size;
    const float* x     = (const float*)d_in[0];
    const float* cheb  = (const float*)d_in[1];
    const float* att   = (const float*)d_in[2];
    const float* theta = (const float*)d_in[3];
    float* out = (float*)d_out;

    dim3 grid(N_ / NTILE, B_);          // 32 x 32 = 1024 workgroups of 512 threads
    cheb_attn_fused<<<grid, THREADS, 0, stream>>>(x, cheb, att, theta, out);
}